// GenderClassifier_35983236006402
// MI455X (gfx1250) — compile-verified
//
#include <hip/hip_runtime.h>

// ---------------------------------------------------------------------------
// Per-sample gathered matvec on MI455X (gfx1250, wave32).
//
// out[b,o] = sum_i x[b,i] * W[(980 - t[b])/20, o, i] + bias[idx, o]
//
// Roofline: 268 MB of x @ 23.3 TB/s ~= 11.5 us floor; W (6.5 MB) is
// L2-resident (192 MB L2, ~82x reuse/row). MACs ride the matrix pipe via
// V_WMMA_F32_16X16X4_F32: one wave embeds 16 per-sample dot products in the
// tile diagonal (A row m = x sample m, B col n = sample n's gathered W row),
// so the XDL unit co-executes with VMEM addressing. Executed matrix FLOPs
// (~4.3 GF) sit far below the bandwidth floor, so 1/16 tile utilization is
// free.
//
// v2 fixes: wave-uniform k-loop (khalf folded into base pointers, not the
// induction variable) so the inner loop fully unrolls with immediate offsets
// and a scalar exit branch (EXEC stays all-1s around WMMA by construction);
// prefetch locality raised to near-cache scope.
// ---------------------------------------------------------------------------

typedef float v2f __attribute__((ext_vector_type(2)));
typedef float v8f __attribute__((ext_vector_type(8)));

#define IN_DIM        16384
#define OUT_CH        2
#define BATCH         4096
#define SAMPLES_PER_WG 16
#define WAVES_PER_WG   8
#define THREADS_PER_WG (32 * WAVES_PER_WG)          // 256
#define K_PER_WAVE    (IN_DIM / WAVES_PER_WG)       // 2048
#define K_OUTER       64                            // floats per unrolled group

__global__ __launch_bounds__(THREADS_PER_WG)
void diag_wmma_gathered_matvec(const float* __restrict__ x,
                               const int*   __restrict__ t,
                               const float* __restrict__ W,
                               const float* __restrict__ bias,
                               float*       __restrict__ out) {
    // per-wave partial diagonals: [wave][channel][sample-in-tile]
    __shared__ float partial[WAVES_PER_WG][OUT_CH][SAMPLES_PER_WG];

    const int tid   = threadIdx.x;
    const int wave  = tid >> 5;
    const int lane  = tid & 31;
    const int sbase = blockIdx.x * SAMPLES_PER_WG;

    // Fragment geometry (CDNA5 ISA, 32-bit 16x4 A layout):
    //   lanes  0-15: M = lane,    VGPR0=K0, VGPR1=K1
    //   lanes 16-31: M = lane-16, VGPR0=K2, VGPR1=K3
    // B (4x16) mirrors: lanes 0-15 hold rows K0/K1 for col N=lane,
    //   lanes 16-31 hold rows K2/K3 for col N=lane-16.
    const int n     = lane & 15;             // sample index within tile
    const int khalf = (lane >> 4) << 1;      // 0 or 2 (per-lane, address only)

    // Per-lane gathered weight row (W is L2-resident).
    const int tn   = t[sbase + n];
    const int widx = (980 - tn) / 20;        // 0..49

    // Per-lane base pointers; khalf lives here, NOT in the loop bounds.
    const float* pa  = x + (size_t)(sbase + n) * IN_DIM
                         + (size_t)wave * K_PER_WAVE + khalf;
    const float* pb0 = W + ((size_t)widx * OUT_CH + 0) * IN_DIM
                         + (size_t)wave * K_PER_WAVE + khalf;
    const float* pb1 = W + ((size_t)widx * OUT_CH + 1) * IN_DIM
                         + (size_t)wave * K_PER_WAVE + khalf;

    v8f c0 = {};   // channel 0 accumulator tile
    v8f c1 = {};   // channel 1 accumulator tile

    // Wave-uniform trip count -> scalar exit branch, EXEC all-1s throughout.
    for (int kt = 0; kt < K_PER_WAVE; kt += K_OUTER) {
        // Stay one group ahead on the HBM x stream (near-cache locality).
        __builtin_prefetch(pa + K_OUTER, 0, 3);
#pragma unroll
        for (int u = 0; u < K_OUTER; u += 4) {     // 16 k-tiles per group
            v2f a  = *(const v2f*)(pa  + u);       // global_load_b64, imm offset
            v2f b0 = *(const v2f*)(pb0 + u);       // L2 hit
            v2f b1 = *(const v2f*)(pb1 + u);       // L2 hit
            c0 = __builtin_amdgcn_wmma_f32_16x16x4_f32(
                     false, a, false, b0, (short)0, c0, false, false);
            c1 = __builtin_amdgcn_wmma_f32_16x16x4_f32(
                     false, a, false, b1, (short)0, c1, false, false);
        }
        pa  += K_OUTER;
        pb0 += K_OUTER;
        pb1 += K_OUTER;
    }

    // Diagonal extraction. C/D 16x16 f32 layout:
    //   VGPR v, lanes 0-15 : (M=v,   N=lane)
    //   VGPR v, lanes 16-31: (M=v+8, N=lane-16)
    // diag(m,m): m=0..7 -> lane m, elem m ; m=8..15 -> lane m+16, elem m-8.
    int m = -1, e = 0;
    if (lane < 8)        { m = lane;      e = lane;      }
    else if (lane >= 24) { m = lane - 16; e = lane - 24; }

    float d0 = 0.0f, d1 = 0.0f;
#pragma unroll
    for (int i = 0; i < 8; ++i) {
        d0 = (e == i) ? c0[i] : d0;
        d1 = (e == i) ? c1[i] : d1;
    }
    if (m >= 0) {
        partial[wave][0][m] = d0;
        partial[wave][1][m] = d1;
    }
    __syncthreads();

    // Cross-wave reduction + bias + coalesced store (32 outputs per WG).
    if (tid < SAMPLES_PER_WG * OUT_CH) {
        const int mm = tid >> 1;     // sample in tile
        const int ch = tid & 1;      // channel
        float s = 0.0f;
#pragma unroll
        for (int w = 0; w < WAVES_PER_WG; ++w) s += partial[w][ch][mm];
        const int wi = (980 - t[sbase + mm]) / 20;
        out[(size_t)(sbase + mm) * OUT_CH + ch] = s + bias[wi * OUT_CH + ch];
    }
}

extern "C" void kernel_launch(void* const* d_in, const int* in_sizes, int n_in,
                              void* d_out, int out_size, void* d_ws, size_t ws_size,
                              hipStream_t stream) {
    const float* x    = (const float*)d_in[0];   // [4096,4,64,64] f32
    const int*   t    = (const int*)  d_in[1];   // [4096] timesteps (980..0 step -20)
    const float* W    = (const float*)d_in[2];   // [50,2,16384] f32
    const float* bias = (const float*)d_in[3];   // [50,2] f32
    float*       out  = (float*)d_out;           // [4096,2] f32

    dim3 grid(BATCH / SAMPLES_PER_WG);           // 256 workgroups
    dim3 block(THREADS_PER_WG);                  // 8 waves (wave32)
    diag_wmma_gathered_matvec<<<grid, block, 0, stream>>>(x, t, W, bias, out);
}